// GCNs_8538394985169
// MI455X (gfx1250) — compile-verified
//
#include <hip/hip_runtime.h>
#include <math.h>

typedef float v2f __attribute__((ext_vector_type(2)));
typedef float v8f __attribute__((ext_vector_type(8)));

// ---------------- graph normalization (computed once, reused for all 5 convs) ---------------

__global__ void k_deg_init(float* __restrict__ deg, int n) {
  int i = blockIdx.x * blockDim.x + threadIdx.x;
  if (i < n) deg[i] = 1.0f;  // self-loop weight (PyG default)
}

__global__ void k_deg_accum(const int* __restrict__ dst, const float* __restrict__ ew,
                            float* __restrict__ deg, int e) {
  int i = blockIdx.x * blockDim.x + threadIdx.x;
  if (i < e) atomicAdd(&deg[dst[i]], ew[i]);
}

__global__ void k_dinv(float* __restrict__ deg, int n) {
  int i = blockIdx.x * blockDim.x + threadIdx.x;
  if (i < n) {
    float d = deg[i];
    deg[i] = (d > 0.0f) ? rsqrtf(d) : 0.0f;  // in-place: deg buffer becomes dinv
  }
}

__global__ void k_edge_norm(const int* __restrict__ src, const int* __restrict__ dst,
                            const float* __restrict__ ew, const float* __restrict__ dinv,
                            float* __restrict__ norm, int e) {
  int i = blockIdx.x * blockDim.x + threadIdx.x;
  if (i < e) norm[i] = dinv[src[i]] * ew[i] * dinv[dst[i]];
}

// ---------------- dense GEMM via V_WMMA_F32_16X16X4_F32 -------------------------------------
// out[n, fout] = h[n, fin] @ w[fin, fout]    (n % 16 == 0, fin % 4 == 0, fout % 16 == 0)
// One wave per 16x16 output tile; K stepped by 4.
// A frag (16x4 f32, 2 VGPR): lanes 0-15 hold row M=lane, K={k,k+1}; lanes 16-31 K={k+2,k+3}.
// B frag (4x16 f32, 2 VGPR): lanes 0-15 hold N=lane of rows K={k,k+1}; lanes 16-31 rows {k+2,k+3}.
// C/D (16x16 f32, 8 VGPR): VGPR j -> M = j (lanes 0-15) / j+8 (lanes 16-31), N = lane%16.
__global__ void k_gemm_wmma_f32(const float* __restrict__ h, const float* __restrict__ w,
                                float* __restrict__ out, int n, int fin, int fout) {
  const int waves_per_blk = blockDim.x >> 5;
  const int gwave = blockIdx.x * waves_per_blk + (threadIdx.x >> 5);
  const int nct = fout >> 4;                 // 16-col tiles
  const int nrt = n >> 4;                    // 16-row tiles
  const int rt = gwave / nct;
  const int ct = gwave % nct;
  if (rt >= nrt) return;                     // wave-uniform: EXEC stays all-ones for WMMA

  const int lane = threadIdx.x & 31;
  const int half = lane >> 4;                // 0: lanes 0-15, 1: lanes 16-31
  const int lm   = lane & 15;

  const int arow = (rt << 4) + lm;           // A row owned by this lane
  const int col  = (ct << 4) + lm;           // B/D column owned by this lane
  const float* __restrict__ hrow = h + (size_t)arow * fin + 2 * half;

  v8f c = {};
  for (int k = 0; k < fin; k += 4) {
    v2f a = *(const v2f*)(hrow + k);                       // global_load_b64, coalesced
    const int kb = k + 2 * half;
    v2f b;
    b.x = w[(size_t)kb * fout + col];
    b.y = w[(size_t)(kb + 1) * fout + col];
    // (neg_a, A, neg_b, B, c_mod, C, reuse_a, reuse_b)
    c = __builtin_amdgcn_wmma_f32_16x16x4_f32(false, a, false, b, (short)0, c, false, false);
  }

#pragma unroll
  for (int j = 0; j < 8; ++j) {
    const int r = (rt << 4) + j + 8 * half;
    out[(size_t)r * fout + col] = c[j];
  }
}

// ---------------- aggregation (vectorized float4 path) -------------------------------------

// acc[i, f4] = bias[f4] + hw[i, f4] * dinv[i]^2   (self-loop term; zero-inits accumulator)
__global__ void k_agg_init_v4(const float4* __restrict__ hw, const float* __restrict__ dinv,
                              const float4* __restrict__ bias, float4* __restrict__ acc,
                              int n, int f4) {
  int t = blockIdx.x * blockDim.x + threadIdx.x;
  if (t < n * f4) {
    int i = t / f4, c = t - i * f4;
    float di = dinv[i];
    float s = di * di;
    float4 h = hw[t], b = bias[c], r;
    r.x = b.x + h.x * s;
    r.y = b.y + h.y * s;
    r.z = b.z + h.z * s;
    r.w = b.w + h.w * s;
    acc[t] = r;
  }
}

// acc[dst, f] += hw[src, f] * norm[e]
// f4 = f/4 threads per edge: one global_load_b128 gather + 4 f32 atomics per thread.
// Dominant traffic; consecutive lanes hit consecutive addresses -> coalesced L2 atomics.
__global__ void k_agg_edges_v4(const float4* __restrict__ hw, const float* __restrict__ norm,
                               const int* __restrict__ src, const int* __restrict__ dst,
                               float* __restrict__ acc, int e, int f, int f4) {
  int t = blockIdx.x * blockDim.x + threadIdx.x;
  if (t < e * f4) {
    int ed = t / f4, q = t - ed * f4;
    float w = norm[ed];
    float4 v = hw[(size_t)src[ed] * f4 + q];     // global_load_b128 gather
    float* a = acc + (size_t)dst[ed] * f + q * 4;
    atomicAdd(a + 0, v.x * w);
    atomicAdd(a + 1, v.y * w);
    atomicAdd(a + 2, v.z * w);
    atomicAdd(a + 3, v.w * w);
  }
}

__global__ void k_elu_v4(float4* __restrict__ x, int sz4) {
  int i = blockIdx.x * blockDim.x + threadIdx.x;
  if (i < sz4) {
    float4 v = x[i];
    v.x = (v.x > 0.0f) ? v.x : expm1f(v.x);
    v.y = (v.y > 0.0f) ? v.y : expm1f(v.y);
    v.z = (v.z > 0.0f) ? v.z : expm1f(v.z);
    v.w = (v.w > 0.0f) ? v.w : expm1f(v.w);
    x[i] = v;
  }
}

__global__ void k_bias_relu_v4(const float4* __restrict__ hw, const float4* __restrict__ bias,
                               float4* __restrict__ out, int n, int f4) {
  int t = blockIdx.x * blockDim.x + threadIdx.x;
  if (t < n * f4) {
    int c = t % f4;
    float4 h = hw[t], b = bias[c], r;
    r.x = fmaxf(h.x + b.x, 0.0f);
    r.y = fmaxf(h.y + b.y, 0.0f);
    r.z = fmaxf(h.z + b.z, 0.0f);
    r.w = fmaxf(h.w + b.w, 0.0f);
    out[t] = r;
  }
}

// ---------------- orchestration ------------------------------------------------------------

static inline int cdiv(long long a, int b) { return (int)((a + b - 1) / b); }

extern "C" void kernel_launch(void* const* d_in, const int* in_sizes, int n_in,
                              void* d_out, int out_size, void* d_ws, size_t ws_size,
                              hipStream_t stream) {
  const float* x   = (const float*)d_in[0];   // [n, in_dim]
  const int*   ei  = (const int*)  d_in[1];   // [2, e]
  const float* ew  = (const float*)d_in[2];   // [e]
  const float* W1  = (const float*)d_in[3];   // [in_dim, emb]
  const float* b1  = (const float*)d_in[4];   // [emb]
  const float* Wh  = (const float*)d_in[5];   // [nhid, emb, emb]
  const float* bh  = (const float*)d_in[6];   // [nhid, emb]
  const float* Wl  = (const float*)d_in[7];   // [emb, out_dim]
  const float* bl  = (const float*)d_in[8];   // [out_dim]

  const int emb     = in_sizes[4];
  const int in_dim  = in_sizes[3] / emb;
  const int out_dim = in_sizes[8];
  const int n       = in_sizes[0] / in_dim;
  const int e       = in_sizes[2];
  const int nhid    = in_sizes[6] / emb;
  const int* src = ei;
  const int* dst = ei + e;

  // workspace carve-up (~36.25 MB for the reference sizes); all offsets 16B-aligned
  float* dinv = (float*)d_ws;                         // [n]       (deg -> dinv in place)
  float* norm = dinv + n;                             // [e]
  float* hw   = norm + e;                             // [n, emb]
  float* hbuf = hw + (size_t)n * emb;                 // [n, emb]

  const int B = 256;
  const int emb4 = emb >> 2, out4 = out_dim >> 2;

  // ---- graph normalization (once) ----
  k_deg_init <<<cdiv(n, B), B, 0, stream>>>(dinv, n);
  k_deg_accum<<<cdiv(e, B), B, 0, stream>>>(dst, ew, dinv, e);
  k_dinv     <<<cdiv(n, B), B, 0, stream>>>(dinv, n);
  k_edge_norm<<<cdiv(e, B), B, 0, stream>>>(src, dst, ew, dinv, norm, e);

  // ---- layer 1: x[n,in_dim] -> hbuf[n,emb] ----
  {
    const int waves = (n >> 4) * (emb >> 4);
    k_gemm_wmma_f32<<<cdiv(waves, B / 32), B, 0, stream>>>(x, W1, hw, n, in_dim, emb);
    k_agg_init_v4 <<<cdiv((long long)n * emb4, B), B, 0, stream>>>(
        (const float4*)hw, dinv, (const float4*)b1, (float4*)hbuf, n, emb4);
    k_agg_edges_v4<<<cdiv((long long)e * emb4, B), B, 0, stream>>>(
        (const float4*)hw, norm, src, dst, hbuf, e, emb, emb4);
    k_elu_v4      <<<cdiv((long long)n * emb4, B), B, 0, stream>>>((float4*)hbuf, n * emb4);
  }

  // ---- hidden layers: hbuf -> hbuf ----
  for (int i = 0; i < nhid; ++i) {
    const float* Wi = Wh + (size_t)i * emb * emb;
    const float* bi = bh + (size_t)i * emb;
    const int waves = (n >> 4) * (emb >> 4);
    k_gemm_wmma_f32<<<cdiv(waves, B / 32), B, 0, stream>>>(hbuf, Wi, hw, n, emb, emb);
    k_agg_init_v4 <<<cdiv((long long)n * emb4, B), B, 0, stream>>>(
        (const float4*)hw, dinv, (const float4*)bi, (float4*)hbuf, n, emb4);
    k_agg_edges_v4<<<cdiv((long long)e * emb4, B), B, 0, stream>>>(
        (const float4*)hw, norm, src, dst, hbuf, e, emb, emb4);
    k_elu_v4      <<<cdiv((long long)n * emb4, B), B, 0, stream>>>((float4*)hbuf, n * emb4);
  }

  // ---- final linear + relu -> d_out ----
  {
    const int waves = (n >> 4) * (out_dim >> 4);
    k_gemm_wmma_f32<<<cdiv(waves, B / 32), B, 0, stream>>>(hbuf, Wl, hw, n, emb, out_dim);
    k_bias_relu_v4<<<cdiv((long long)n * out4, B), B, 0, stream>>>(
        (const float4*)hw, (const float4*)bl, (float4*)d_out, n, out4);
  }
}